// HeterogeneousSelfAttention_69827578298376
// MI455X (gfx1250) — compile-verified
//
#include <hip/hip_runtime.h>
#include <hip/hip_bf16.h>

// ---------------------------------------------------------------------------
// Problem constants (match reference)
// ---------------------------------------------------------------------------
constexpr int B_    = 4;
constexpr int N_    = 1024;
constexpr int HID_  = 768;
constexpr int NH_   = 12;
constexpr int DH_   = 64;
constexpr int E_    = 65536;
constexpr int BN_   = B_ * N_;          // 4096 rows

typedef float v2f __attribute__((ext_vector_type(2)));
typedef float v8f __attribute__((ext_vector_type(8)));

// V_WMMA_F32_16X16X4_F32 : D(16x16 f32) = A(16x4 f32) * B(4x16 f32) + C
// A,B = 2 VGPRs each; C/D = 8 VGPRs. 8-arg form per probe doc.
__device__ __forceinline__ v8f wmma_f32_16x16x4(v2f a, v2f b, v8f c) {
    return __builtin_amdgcn_wmma_f32_16x16x4_f32(
        /*neg_a=*/false, a, /*neg_b=*/false, b,
        /*c_mod=*/(short)0, c, /*reuse_a=*/false, /*reuse_b=*/false);
}

// monotone float -> uint mapping (order preserving) for atomicMax-based segmax
__device__ __forceinline__ unsigned f32_mono(float f) {
    unsigned u = __float_as_uint(f);
    return (u & 0x80000000u) ? ~u : (u | 0x80000000u);
}
__device__ __forceinline__ float f32_demono(unsigned u) {
    unsigned b = (u & 0x80000000u) ? (u & 0x7FFFFFFFu) : ~u;
    return __uint_as_float(b);
}

// ---------------------------------------------------------------------------
// Kernel 1: fused dual projection  Out[n,:] = tok(n) ? X@Wt^T+bt : X@Wk^T+bk
// One wave computes a 16x64 output tile, two accumulator sets (tok / kb)
// sharing the A operand of every WMMA.
// ---------------------------------------------------------------------------
__global__ __launch_bounds__(256)
void dual_proj_kernel(const float* __restrict__ X,        // (4096, 768)
                      const int*   __restrict__ ntype,    // (4096)
                      const float* __restrict__ Wt,       // (768, 768)
                      const float* __restrict__ bt,       // (768)
                      const float* __restrict__ Wkb,      // (768, 768)
                      const float* __restrict__ bkb,      // (768)
                      float* __restrict__ Out)            // (4096, 768)
{
    const int wave   = blockIdx.x * 8 + (threadIdx.x >> 5);
    const int lane   = threadIdx.x & 31;
    const int half   = lane >> 4;        // 0: K0/K1, rows 0..7 | 1: K2/K3, rows 8..15
    const int lr     = lane & 15;

    const int rowTile = wave / 12;       // 256 row tiles of 16
    const int cg      = wave % 12;       // 12 column groups of 64
    const int row0    = rowTile * 16;
    const int col0    = cg * 64;

    v8f at[4] = {v8f{}, v8f{}, v8f{}, v8f{}};   // tok accumulators (4 N-tiles)
    v8f ak[4] = {v8f{}, v8f{}, v8f{}, v8f{}};   // kb  accumulators

    const float* xrow = X + (size_t)(row0 + lr) * HID_ + 2 * half;

    for (int kk = 0; kk < HID_; kk += 4) {
        v2f a = *(const v2f*)(xrow + kk);
#pragma unroll
        for (int nt = 0; nt < 4; ++nt) {
            const int j = col0 + nt * 16 + lr;
            const float* wt = Wt  + (size_t)j * HID_ + kk + 2 * half;
            const float* wk = Wkb + (size_t)j * HID_ + kk + 2 * half;
            v2f btile = *(const v2f*)wt;   // W[j,kk+s], W[j,kk+s+1] consecutive
            v2f ktile = *(const v2f*)wk;
            at[nt] = wmma_f32_16x16x4(a, btile, at[nt]);
            ak[nt] = wmma_f32_16x16x4(a, ktile, ak[nt]);
        }
    }

    // epilogue: bias + per-row token/kb select, store
    float btv[4], bkv[4];
#pragma unroll
    for (int nt = 0; nt < 4; ++nt) {
        const int j = col0 + nt * 16 + lr;
        btv[nt] = bt[j];
        bkv[nt] = bkb[j];
    }
#pragma unroll
    for (int i = 0; i < 8; ++i) {
        const int m = row0 + i + 8 * half;         // C layout: row = i + 8*(lane>=16)
        const bool istok = (ntype[m] == 0);
        float* orow = Out + (size_t)m * HID_;
#pragma unroll
        for (int nt = 0; nt < 4; ++nt) {
            const int j = col0 + nt * 16 + lr;
            orow[j] = istok ? (at[nt][i] + btv[nt]) : (ak[nt][i] + bkv[nt]);
        }
    }
}

// ---------------------------------------------------------------------------
// Kernel 2: per-edge logits.  One wave per edge.
//   QM(16x64) = Qe(16x64, heads padded) @ M[er](64x64)   via 64 WMMAs
//   logits[h] = 0.125 * sum_f QM[h,f] * K[h,f]
// ---------------------------------------------------------------------------
__global__ __launch_bounds__(256)
void edge_logits_kernel(const float* __restrict__ Q,
                        const float* __restrict__ K,
                        const int*   __restrict__ eidx,    // (4, E)
                        const float* __restrict__ relmats, // (64, 64, 64)
                        float* __restrict__ logits)        // (E, 12)
{
    const int e    = blockIdx.x * 8 + (threadIdx.x >> 5);
    const int lane = threadIdx.x & 31;
    const int half = lane >> 4;
    const int lr   = lane & 15;

    const int eb = eidx[e];
    const int eh = eidx[E_ + e];
    const int et = eidx[2 * E_ + e];
    const int er = eidx[3 * E_ + e];

    const size_t qb = (size_t)(eb * N_ + eh) * HID_;
    const size_t kb = (size_t)(eb * N_ + et) * HID_;
    const size_t mb = (size_t)er * (DH_ * DH_);

    v8f acc[4] = {v8f{}, v8f{}, v8f{}, v8f{}};

    const int hld = (lr < NH_) ? lr : (NH_ - 1);   // clamp padded head rows in-bounds
    const float* qp = Q + qb + hld * DH_ + 2 * half;

    for (int kk = 0; kk < DH_; kk += 4) {
        v2f a = *(const v2f*)(qp + kk);
#pragma unroll
        for (int nt = 0; nt < 4; ++nt) {
            const int f = nt * 16 + lr;
            const float* mp = relmats + mb + (size_t)(kk + 2 * half) * DH_ + f;
            v2f b;
            b.x = mp[0];        // M[kk+2h  , f]
            b.y = mp[DH_];      // M[kk+2h+1, f]
            acc[nt] = wmma_f32_16x16x4(a, b, acc[nt]);
        }
    }

    // dot with K along f, per head, then shuffle-reduce across the 16 lanes
    float part[8];
#pragma unroll
    for (int i = 0; i < 8; ++i) part[i] = 0.0f;
#pragma unroll
    for (int i = 0; i < 8; ++i) {
        const int head = i + 8 * half;
        if (head < NH_) {
            const float* krow = K + kb + (size_t)head * DH_;
#pragma unroll
            for (int nt = 0; nt < 4; ++nt) {
                const int f = nt * 16 + lr;
                part[i] += acc[nt][i] * krow[f];
            }
        }
    }
#pragma unroll
    for (int i = 0; i < 8; ++i) {
        part[i] += __shfl_xor(part[i], 8);
        part[i] += __shfl_xor(part[i], 4);
        part[i] += __shfl_xor(part[i], 2);
        part[i] += __shfl_xor(part[i], 1);
    }
    if (lr == 0) {
#pragma unroll
        for (int i = 0; i < 8; ++i) {
            const int head = i + 8 * half;
            if (head < NH_) logits[(size_t)e * NH_ + head] = part[i] * 0.125f;
        }
    }
}

// ---------------------------------------------------------------------------
// Kernel 3: segment max via monotone-uint atomicMax
// ---------------------------------------------------------------------------
__global__ __launch_bounds__(256)
void seg_max_kernel(const float* __restrict__ logits,
                    const int*   __restrict__ eidx,
                    unsigned*    __restrict__ segmax)     // (B*N*NH)
{
    const int t = blockIdx.x * 256 + threadIdx.x;
    if (t >= E_ * NH_) return;
    const int e = t / NH_, h = t - e * NH_;
    const int eb = eidx[e], eh = eidx[E_ + e];
    atomicMax(&segmax[(size_t)(eb * N_ + eh) * NH_ + h], f32_mono(logits[t]));
}

// ---------------------------------------------------------------------------
// Kernel 4: exp(logit - max) in place + segment sum
// ---------------------------------------------------------------------------
__global__ __launch_bounds__(256)
void seg_exp_sum_kernel(float* __restrict__ logits,           // in/out: -> ex
                        const int* __restrict__ eidx,
                        const unsigned* __restrict__ segmax,
                        float* __restrict__ segsum)           // (B*N*NH)
{
    const int t = blockIdx.x * 256 + threadIdx.x;
    if (t >= E_ * NH_) return;
    const int e = t / NH_, h = t - e * NH_;
    const int eb = eidx[e], eh = eidx[E_ + e];
    const size_t s = (size_t)(eb * N_ + eh) * NH_ + h;
    const float mx = f32_demono(segmax[s]);
    const float ex = __expf(logits[t] - mx);
    logits[t] = ex;
    unsafeAtomicAdd(&segsum[s], ex);
}

// ---------------------------------------------------------------------------
// Kernel 5: message scatter.  One wave per edge, 768 floats/edge,
// native global_atomic_add_f32 into the output.
// ---------------------------------------------------------------------------
__global__ __launch_bounds__(256)
void scatter_kernel(const float* __restrict__ V,
                    const float* __restrict__ edge_emb,  // (64, 768)
                    const int*   __restrict__ eidx,
                    const float* __restrict__ probs,     // (E, 12) = ex
                    const float* __restrict__ segsum,
                    float* __restrict__ Out)             // (4096, 768)
{
    const int e    = blockIdx.x * 8 + (threadIdx.x >> 5);
    const int lane = threadIdx.x & 31;

    const int eb = eidx[e];
    const int eh = eidx[E_ + e];
    const int et = eidx[2 * E_ + e];
    const int er = eidx[3 * E_ + e];

    const size_t obase = (size_t)(eb * N_ + eh) * HID_;
    const size_t vbase = (size_t)(eb * N_ + et) * HID_;
    const size_t ebase = (size_t)er * HID_;
    const size_t pbase = (size_t)e * NH_;
    const size_t sbase = (size_t)(eb * N_ + eh) * NH_;

#pragma unroll
    for (int j = lane; j < HID_; j += 32) {
        const int h = j >> 6;                           // head = j / 64
        const float p = probs[pbase + h] / segsum[sbase + h];
        const float val = V[vbase + j] + edge_emb[ebase + j];
        unsafeAtomicAdd(&Out[obase + j], p * val);
    }
}

// ---------------------------------------------------------------------------
// Host-side launch
// ---------------------------------------------------------------------------
extern "C" void kernel_launch(void* const* d_in, const int* in_sizes, int n_in,
                              void* d_out, int out_size, void* d_ws, size_t ws_size,
                              hipStream_t stream)
{
    const float* node_states = (const float*)d_in[0];
    const int*   eidx        = (const int*)  d_in[1];
    const int*   ntype       = (const int*)  d_in[2];
    const float* Wq    = (const float*)d_in[3];
    const float* bq    = (const float*)d_in[4];
    const float* Wk    = (const float*)d_in[5];
    const float* bk    = (const float*)d_in[6];
    const float* Wv    = (const float*)d_in[7];
    const float* bv    = (const float*)d_in[8];
    const float* Wq_kb = (const float*)d_in[9];
    const float* bq_kb = (const float*)d_in[10];
    const float* Wk_kb = (const float*)d_in[11];
    const float* bk_kb = (const float*)d_in[12];
    const float* Wv_kb = (const float*)d_in[13];
    const float* bv_kb = (const float*)d_in[14];
    const float* relmats  = (const float*)d_in[15];
    const float* edge_emb = (const float*)d_in[16];
    float* out = (float*)d_out;

    // workspace layout (floats)
    float*    Qb   = (float*)d_ws;
    float*    Kb   = Qb + (size_t)BN_ * HID_;          // 3,145,728 each
    float*    Vb   = Kb + (size_t)BN_ * HID_;
    float*    Lg   = Vb + (size_t)BN_ * HID_;          // E*NH = 786,432
    unsigned* SMx  = (unsigned*)(Lg + (size_t)E_ * NH_);   // 49,152 u32
    float*    SSum = (float*)(SMx + (size_t)BN_ * NH_);    // 49,152 f32

    // zero output + softmax accumulators (segmax monotone code 0 == -inf)
    hipMemsetAsync(out, 0, (size_t)out_size * sizeof(float), stream);
    hipMemsetAsync(SMx, 0, (size_t)BN_ * NH_ * 2 * sizeof(unsigned), stream);

    // 1) dual projections: 3072 wave-tiles / 8 waves per block
    dim3 blk(256);
    dim3 gproj(384);
    dual_proj_kernel<<<gproj, blk, 0, stream>>>(node_states, ntype, Wq, bq, Wq_kb, bq_kb, Qb);
    dual_proj_kernel<<<gproj, blk, 0, stream>>>(node_states, ntype, Wk, bk, Wk_kb, bk_kb, Kb);
    dual_proj_kernel<<<gproj, blk, 0, stream>>>(node_states, ntype, Wv, bv, Wv_kb, bv_kb, Vb);

    // 2) per-edge logits (one wave per edge)
    edge_logits_kernel<<<dim3(E_ / 8), blk, 0, stream>>>(Qb, Kb, eidx, relmats, Lg);

    // 3) segment max
    seg_max_kernel<<<dim3((E_ * NH_ + 255) / 256), blk, 0, stream>>>(Lg, eidx, SMx);

    // 4) exp + segment sum
    seg_exp_sum_kernel<<<dim3((E_ * NH_ + 255) / 256), blk, 0, stream>>>(Lg, eidx, SMx, SSum);

    // 5) weighted scatter of values into output
    scatter_kernel<<<dim3(E_ / 8), blk, 0, stream>>>(Vb, edge_emb, eidx, Lg, SSum, out);
}